// Router_17394617549052
// MI455X (gfx1250) — compile-verified
//
#include <hip/hip_runtime.h>
#include <hip/hip_bf16.h>
#include <cmath>

typedef float v2f __attribute__((ext_vector_type(2)));
typedef float v8f __attribute__((ext_vector_type(8)));

// Problem constants (fixed by the reference)
constexpr int Bc = 8, Sc = 2048, Dc = 2048, Ec = 8;
constexpr int Tc = Bc * Sc;                       // 16384 tokens
constexpr int SPLITK = 4;                         // waves cooperating on one tile
constexpr int WAVES  = 8;                         // waves per block (wave32)
constexpr int BLOCK  = WAVES * 32;                // 256 threads
constexpr int TILES_PER_BLOCK = WAVES / SPLITK;   // 2 tiles of 16 tokens
constexpr int TOK_PER_BLOCK   = TILES_PER_BLOCK * 16;  // 32
constexpr int KSLICE = Dc / SPLITK;               // 512 K-elements per wave
constexpr int KCHUNK = 256;                       // uniform prefetch granularity

__global__ __launch_bounds__(BLOCK)
void router_wmma_kernel(const float* __restrict__ x,
                        const float* __restrict__ gate_w,
                        const float* __restrict__ gate_b,
                        const float* __restrict__ noise_w,
                        const float* __restrict__ noise_b,
                        const float* __restrict__ noise,
                        float* __restrict__ out)
{
    // Partial 16x16 C per (tile, k-slice); pad to 17 to avoid bank conflicts.
    __shared__ float cbuf[TILES_PER_BLOCK][SPLITK][16][17];

    const int tid  = threadIdx.x;
    const int wave = tid >> 5;
    const int lane = tid & 31;
    const int hl   = lane >> 4;   // 0 = lanes 0-15, 1 = lanes 16-31
    const int l16  = lane & 15;

    const int tile = wave / SPLITK;        // which 16-token tile in this block
    const int ks   = wave % SPLITK;        // which K-slice this wave reduces
    const int k0   = ks * KSLICE;

    const int tokenBase = blockIdx.x * TOK_PER_BLOCK + tile * 16;

    // Combined 16-wide weight matrix W = [gate_w | noise_w]; lane owns column l16.
    const float* wcol = (l16 < Ec) ? (gate_w + l16) : (noise_w + (l16 - Ec));
    // This lane's A-matrix row of x (token tokenBase + l16).
    const float* xrow = x + (size_t)(tokenBase + l16) * Dc;

    v8f c = {0.f, 0.f, 0.f, 0.f, 0.f, 0.f, 0.f, 0.f};

    // V_WMMA_F32_16X16X4_F32: A 16x4 f32, B 4x16 f32, C 16x16 f32.
    // A layout: VGPR0 = K0 (lanes 0-15) / K2 (lanes 16-31); VGPR1 = K1 / K3
    //  -> lane loads x[row][k + 2*hl], x[row][k + 2*hl + 1]  (one b64 load).
    // B mirrors A: VGPR0 = row K0/K2 at col l16; VGPR1 = K1/K3.
    //
    // Outer loop: uniform, branch-free-in-inner prefetch of the x stream.
    for (int kc = k0; kc < k0 + KSLICE; kc += KCHUNK) {
        __builtin_prefetch(xrow + kc + KCHUNK, 0, 3);   // next chunk of x
#pragma unroll 8
        for (int k = kc; k < kc + KCHUNK; k += 4) {
            const int kk = k + 2 * hl;
            v2f a = *(const v2f*)(xrow + kk);
            v2f b;
            b.x = wcol[(size_t)kk * Ec];
            b.y = wcol[(size_t)(kk + 1) * Ec];
            c = __builtin_amdgcn_wmma_f32_16x16x4_f32(
                    false, a, false, b, (short)0, c, false, false);
        }
    }

    // C 16x16 f32 layout: VGPR j -> (M=j, N=lane) lanes 0-15; (M=j+8, N=lane-16)
    // lanes 16-31. Deposit partials; transpose through LDS so one lane owns a token.
#pragma unroll
    for (int j = 0; j < 8; ++j)
        cbuf[tile][ks][j + hl * 8][l16] = c[j];
    __syncthreads();

    // ---- Fused epilogue (one wave per tile): reduce split-K partials, softplus
    // noise-std, noisy logits, top-2, sparse softmax. No [B,S,E] HBM intermediate.
    if (ks == 0 && lane < 16) {
        const int t = tokenBase + lane;
        float nl[8];
#pragma unroll
        for (int e = 0; e < 8; ++e) {
            float lg = gate_b[e];
            float r  = noise_b[e];
#pragma unroll
            for (int s = 0; s < SPLITK; ++s) {
                lg += cbuf[tile][s][lane][e];
                r  += cbuf[tile][s][lane][8 + e];
            }
            // numerically stable softplus
            float sp = fmaxf(r, 0.f) + log1pf(__expf(-fabsf(r)));
            nl[e] = lg + noise[(size_t)t * Ec + e] * sp;
        }
        // top-2 (strict > keeps lowest index on ties, matching jax.lax.top_k)
        int i1 = 0; float v1 = nl[0];
#pragma unroll
        for (int e = 1; e < 8; ++e)
            if (nl[e] > v1) { v1 = nl[e]; i1 = e; }
        int i2 = -1; float v2 = -INFINITY;
#pragma unroll
        for (int e = 0; e < 8; ++e)
            if (e != i1 && nl[e] > v2) { v2 = nl[e]; i2 = e; }

        // softmax over the two kept entries (-inf elsewhere -> prob 0)
        float e2  = __expf(v2 - v1);
        float inv = 1.f / (1.f + e2);
#pragma unroll
        for (int e = 0; e < 8; ++e) {
            float p = (e == i1) ? inv : ((e == i2) ? e2 * inv : 0.f);
            out[(size_t)t * Ec + e] = p;
        }
        // topk indices, concatenated after probs (written as float values)
        float* oi = out + (size_t)Tc * Ec;
        oi[(size_t)t * 2 + 0] = (float)i1;
        oi[(size_t)t * 2 + 1] = (float)i2;
    }
}

extern "C" void kernel_launch(void* const* d_in, const int* in_sizes, int n_in,
                              void* d_out, int out_size, void* d_ws, size_t ws_size,
                              hipStream_t stream) {
    const float* x       = (const float*)d_in[0];
    const float* gate_w  = (const float*)d_in[1];
    const float* gate_b  = (const float*)d_in[2];
    const float* noise_w = (const float*)d_in[3];
    const float* noise_b = (const float*)d_in[4];
    const float* noise   = (const float*)d_in[5];
    // d_in[6] = top_k (fixed at 2 in this specialization)
    float* out = (float*)d_out;

    dim3 grid(Tc / TOK_PER_BLOCK);   // 512 blocks
    dim3 block(BLOCK);               // 256 threads = 8 waves
    router_wmma_kernel<<<grid, block, 0, stream>>>(
        x, gate_w, gate_b, noise_w, noise_b, noise, out);
}